// ExerciseGNN_77171972374635
// MI455X (gfx1250) — compile-verified
//
#include <hip/hip_runtime.h>

#define N_NODES   100000
#define N_EDGES   1600000
#define N_FEAT    66
#define K_PAD     68
#define HID       128
#define N_CLASSES 3
#define N_GRAPHS  256

typedef __attribute__((ext_vector_type(2))) float v2f;
typedef __attribute__((ext_vector_type(8))) float v8f;

// ---------------------------------------------------------------- utilities
__global__ void fill_f32(float* __restrict__ p, float v, int n) {
    int i = blockIdx.x * blockDim.x + threadIdx.x;
    if (i < n) p[i] = v;
}

// pad x [N,66] -> xp [N,68] (zeros in cols 66..67)
__global__ void pad_x_kernel(const float* __restrict__ x, float* __restrict__ xp) {
    int gid = blockIdx.x * blockDim.x + threadIdx.x;
    if (gid < N_NODES * K_PAD) {
        const int i = gid / K_PAD;
        const int c = gid % K_PAD;
        xp[gid] = (c < N_FEAT) ? x[(size_t)i * N_FEAT + c] : 0.0f;
    }
}

// pad W1 [66,128] -> wp [68,128] (zero rows 66..67)
__global__ void pad_w_kernel(const float* __restrict__ w, float* __restrict__ wp) {
    int gid = blockIdx.x * blockDim.x + threadIdx.x;
    if (gid < K_PAD * HID)
        wp[gid] = (gid < N_FEAT * HID) ? w[gid] : 0.0f;
}

// deg[dst] += 1 over all edges (accumulated into `dis` buffer)
__global__ void deg_kernel(const int* __restrict__ dst, float* __restrict__ deg) {
    int e = blockIdx.x * blockDim.x + threadIdx.x;
    if (e < N_EDGES) atomicAdd(&deg[dst[e]], 1.0f);
}

// dis = rsqrt(deg + 1)   (self-loop included, PyG gcn_norm)
__global__ void rsqrt_kernel(float* __restrict__ dis) {
    int i = blockIdx.x * blockDim.x + threadIdx.x;
    if (i < N_NODES) dis[i] = rsqrtf(dis[i] + 1.0f);
}

// ------------------------------------------------------- fp32 WMMA GEMM
// C[M x HID] = A[M x K] @ W[K x HID]; M = N_NODES (multiple of 16), K % 4 == 0.
// One wave computes one 16x16 tile via V_WMMA_F32_16X16X4_F32.
// Block = 256 threads = 8 waves -> all 8 column tiles of HID=128.
// A tile staged in LDS (shared by all 8 waves); +2 row pad breaks bank
// conflicts and keeps 8-byte alignment for float2 fragment loads (k0 even).
template <int K>
__global__ __launch_bounds__(256)
void gemm_wmma_f32(const float* __restrict__ A, const float* __restrict__ W,
                   float* __restrict__ C) {
    static_assert(K % 4 == 0, "K must be a multiple of 4");
    __shared__ float As[16][K + 2];

    const int tid  = threadIdx.x;
    const int lane = tid & 31;
    const int wave = tid >> 5;
    const int row0 = blockIdx.x << 4;   // M tile

    // cooperative A-tile load: 16*K floats across 256 threads (coalesced rows)
    for (int idx = tid; idx < 16 * K; idx += 256) {
        const int r = idx / K;
        const int c = idx % K;
        As[r][c] = A[(size_t)(row0 + r) * K + c];
    }
    __syncthreads();

    const int col0 = wave << 4;         // N tile
    const int m    = lane & 15;         // row within A tile / col within B,C tile
    const int kph  = (lane >> 4) << 1;  // lane-half K offset: 0 or 2

    v8f acc = {};
    const float* wcol = W + col0 + m;

    #pragma unroll 4
    for (int k = 0; k < K; k += 4) {
        const int k0 = k + kph;
        // A 16x4 fragment: lanes 0-15 hold K=k+0,k+1; lanes 16-31 hold K=k+2,k+3
        const v2f a = *(const v2f*)(&As[m][k0]);
        // B 4x16 fragment: rows striped across lanes; lane-half selects row pair
        v2f b;
        b.x = wcol[(size_t)k0 * HID];
        b.y = wcol[(size_t)(k0 + 1) * HID];
        acc = __builtin_amdgcn_wmma_f32_16x16x4_f32(
            /*neg_a=*/false, a, /*neg_b=*/false, b,
            /*c_mod=*/(short)0, acc, /*reuse_a=*/false, /*reuse_b=*/false);
    }

    // C/D layout: VGPR i -> row i (lanes 0-15) / row i+8 (lanes 16-31), col = lane&15
    const int rhalf = (lane >> 4) << 3;
    #pragma unroll
    for (int i = 0; i < 8; ++i)
        C[(size_t)(row0 + i + rhalf) * HID + col0 + m] = acc[i];
}

// --------------------------------------------------- edge scatter-aggregate
// One wave per edge: 32 lanes x float4 = 128 features. AGG[dst] += H[src]*norm.
__global__ void edge_agg(const float* __restrict__ H, const float* __restrict__ dis,
                         const int* __restrict__ src, const int* __restrict__ dst,
                         float* __restrict__ AGG) {
    const int tid  = blockIdx.x * blockDim.x + threadIdx.x;
    const int lane = tid & 31;
    const int estr = (gridDim.x * blockDim.x) >> 5;
    for (int e = tid >> 5; e < N_EDGES; e += estr) {
        const int s = src[e];
        const int d = dst[e];
        const float nrm = dis[s] * dis[d];
        const float4 v  = ((const float4*)(H + (size_t)s * HID))[lane];
        float* o = AGG + (size_t)d * HID + lane * 4;
        atomicAdd(o + 0, v.x * nrm);
        atomicAdd(o + 1, v.y * nrm);
        atomicAdd(o + 2, v.z * nrm);
        atomicAdd(o + 3, v.w * nrm);
    }
}

// OUT = AGG + H * dis^2 + bias, optional ReLU
__global__ void combine_kernel(const float* __restrict__ AGG, const float* __restrict__ H,
                               const float* __restrict__ dis, const float* __restrict__ bias,
                               float* __restrict__ OUT, int relu) {
    int gid = blockIdx.x * blockDim.x + threadIdx.x;
    if (gid < N_NODES * HID) {
        const int i = gid >> 7;
        const int f = gid & (HID - 1);
        const float di = dis[i];
        float v = AGG[gid] + H[gid] * di * di + bias[f];
        if (relu) v = fmaxf(v, 0.0f);
        OUT[gid] = v;
    }
}

// ------------------------------------------------------------- pooling
__global__ void pool_cnt(const int* __restrict__ batch, float* __restrict__ cnt) {
    int i = blockIdx.x * blockDim.x + threadIdx.x;
    if (i < N_NODES) atomicAdd(&cnt[batch[i]], 1.0f);
}

__global__ void pool_sum(const float* __restrict__ P, const int* __restrict__ batch,
                         float* __restrict__ pooled) {
    int gid = blockIdx.x * blockDim.x + threadIdx.x;
    if (gid < N_NODES * HID) {
        const int i = gid >> 7;
        const int f = gid & (HID - 1);
        atomicAdd(&pooled[(size_t)batch[i] * HID + f], P[gid]);
    }
}

__global__ void pool_div(float* __restrict__ pooled, const float* __restrict__ cnt) {
    int gid = blockIdx.x * blockDim.x + threadIdx.x;
    if (gid < N_GRAPHS * HID) {
        const int g = gid >> 7;
        pooled[gid] = pooled[gid] / fmaxf(cnt[g], 1.0f);
    }
}

// ------------------------------------------------------------- MLP head
__global__ void mlp1(const float* __restrict__ pooled, const float* __restrict__ W4,
                     const float* __restrict__ b4, float* __restrict__ z) {
    int gid = blockIdx.x * blockDim.x + threadIdx.x;   // 256*64 outputs
    if (gid < N_GRAPHS * (HID / 2)) {
        const int g = gid >> 6;
        const int j = gid & 63;
        float s = b4[j];
        const float* pg = pooled + (size_t)g * HID;
        for (int k = 0; k < HID; ++k) s += pg[k] * W4[(size_t)k * (HID / 2) + j];
        z[gid] = fmaxf(s, 0.0f);
    }
}

__global__ void mlp2(const float* __restrict__ z, const float* __restrict__ W5,
                     const float* __restrict__ b5, float* __restrict__ out) {
    int gid = blockIdx.x * blockDim.x + threadIdx.x;   // 256*3 outputs
    if (gid < N_GRAPHS * N_CLASSES) {
        const int g = gid / N_CLASSES;
        const int c = gid % N_CLASSES;
        float s = b5[c];
        const float* zg = z + (size_t)g * (HID / 2);
        for (int k = 0; k < HID / 2; ++k) s += zg[k] * W5[(size_t)k * N_CLASSES + c];
        out[gid] = s;
    }
}

// ------------------------------------------------------------- host launcher
extern "C" void kernel_launch(void* const* d_in, const int* in_sizes, int n_in,
                              void* d_out, int out_size, void* d_ws, size_t ws_size,
                              hipStream_t stream) {
    (void)in_sizes; (void)n_in; (void)out_size; (void)ws_size;

    const float* x     = (const float*)d_in[0];
    const int*   ei    = (const int*)  d_in[1];
    const int*   batch = (const int*)  d_in[2];
    const float* W1 = (const float*)d_in[3];  const float* b1 = (const float*)d_in[4];
    const float* W2 = (const float*)d_in[5];  const float* b2 = (const float*)d_in[6];
    const float* W3 = (const float*)d_in[7];  const float* b3 = (const float*)d_in[8];
    const float* W4 = (const float*)d_in[9];  const float* b4 = (const float*)d_in[10];
    const float* W5 = (const float*)d_in[11]; const float* b5 = (const float*)d_in[12];
    float* out = (float*)d_out;

    const int* src = ei;
    const int* dst = ei + N_EDGES;

    // workspace layout (floats)
    float* ws     = (float*)d_ws;
    float* dis    = ws;                                   // N_NODES
    float* H      = dis + N_NODES;                        // N_NODES*HID
    float* AGG    = H   + (size_t)N_NODES * HID;          // N_NODES*HID
    float* P      = AGG + (size_t)N_NODES * HID;          // N_NODES*HID
    float* pooled = P   + (size_t)N_NODES * HID;          // N_GRAPHS*HID
    float* cnt    = pooled + N_GRAPHS * HID;              // N_GRAPHS
    float* z      = cnt + N_GRAPHS;                       // N_GRAPHS*HID/2
    float* WP1    = z + N_GRAPHS * (HID / 2);             // K_PAD*HID padded W1
    float* XP     = AGG;  // padded x [N,68] aliases AGG (dead until after GEMM1)

    const int B = 256;
    const int gN    = (N_NODES + B - 1) / B;
    const int gE    = (N_EDGES + B - 1) / B;
    const int gNH   = (N_NODES * HID) / B;    // 50000 exactly
    const int gGemm = N_NODES / 16;           // 6250 exactly

    // degree / normalization
    fill_f32<<<gN, B, 0, stream>>>(dis, 0.0f, N_NODES);
    deg_kernel<<<gE, B, 0, stream>>>(dst, dis);
    rsqrt_kernel<<<gN, B, 0, stream>>>(dis);

    // layer 1: pad x and W1 to K=68, GEMM -> H, aggregate -> P (ReLU)
    pad_x_kernel<<<(N_NODES * K_PAD + B - 1) / B, B, 0, stream>>>(x, XP);
    pad_w_kernel<<<(K_PAD * HID + B - 1) / B, B, 0, stream>>>(W1, WP1);
    gemm_wmma_f32<K_PAD><<<gGemm, B, 0, stream>>>(XP, WP1, H);
    fill_f32<<<gNH, B, 0, stream>>>(AGG, 0.0f, N_NODES * HID);
    edge_agg<<<4096, B, 0, stream>>>(H, dis, src, dst, AGG);
    combine_kernel<<<gNH, B, 0, stream>>>(AGG, H, dis, b1, P, 1);

    // layer 2: P -> P (ReLU)
    gemm_wmma_f32<HID><<<gGemm, B, 0, stream>>>(P, W2, H);
    fill_f32<<<gNH, B, 0, stream>>>(AGG, 0.0f, N_NODES * HID);
    edge_agg<<<4096, B, 0, stream>>>(H, dis, src, dst, AGG);
    combine_kernel<<<gNH, B, 0, stream>>>(AGG, H, dis, b2, P, 1);

    // layer 3: P -> P (no ReLU)
    gemm_wmma_f32<HID><<<gGemm, B, 0, stream>>>(P, W3, H);
    fill_f32<<<gNH, B, 0, stream>>>(AGG, 0.0f, N_NODES * HID);
    edge_agg<<<4096, B, 0, stream>>>(H, dis, src, dst, AGG);
    combine_kernel<<<gNH, B, 0, stream>>>(AGG, H, dis, b3, P, 0);

    // global mean pool
    fill_f32<<<(N_GRAPHS * HID + N_GRAPHS + B - 1) / B, B, 0, stream>>>(
        pooled, 0.0f, N_GRAPHS * HID + N_GRAPHS);   // pooled + cnt contiguous
    pool_cnt<<<gN, B, 0, stream>>>(batch, cnt);
    pool_sum<<<gNH, B, 0, stream>>>(P, batch, pooled);
    pool_div<<<(N_GRAPHS * HID + B - 1) / B, B, 0, stream>>>(pooled, cnt);

    // classifier head
    mlp1<<<(N_GRAPHS * (HID / 2) + B - 1) / B, B, 0, stream>>>(pooled, W4, b4, z);
    mlp2<<<(N_GRAPHS * N_CLASSES + B - 1) / B, B, 0, stream>>>(z, W5, b5, out);
}